// PrototypeMemoryBank_45569603010859
// MI455X (gfx1250) — compile-verified
//
#include <hip/hip_runtime.h>
#include <hip/hip_bf16.h>
#include <stdint.h>

// ---------------------------------------------------------------------------
// PrototypeMemoryBank: segment-mean-by-label + EMA-normalize update.
//
// Roofline: reading z = 131072*256*4B = 134MB -> ~5.8us @ 23.3TB/s HBM.
// Arithmetic intensity ~0.25 FLOP/B => pure bandwidth problem. WMMA (one-hot
// GEMM) would cost 67 GFLOP of matmul for 34 MFLOP of useful adds -> slower
// and lossy. The right gfx1250 path is the async data mover:
//   global_load_async_to_lds_b128 (ASYNCcnt DMA, double-buffered) -> LDS ->
//   native global_atomic_add_f32 scatter into a 1MB L2-resident accumulator.
// ---------------------------------------------------------------------------

#define NUM_CLASSES 1000
#define EMBED_DIM   256
#define MOMENTUM    0.995f
#define EPS_NORM    1e-12f

#define TILE_ROWS       32                       // 32 rows * 1KB = 32KB per tile
#define TILES_PER_BLOCK 8
#define ROWS_PER_BLOCK  (TILE_ROWS * TILES_PER_BLOCK)   // 256
#define THREADS         256

// 128-bit payload type expected by the async-DMA builtin (int4 pointee).
typedef int v4i __attribute__((ext_vector_type(4)));
typedef __attribute__((address_space(1))) v4i* gv4i_p;   // global
typedef __attribute__((address_space(3))) v4i* lv4i_p;   // LDS

// ---- CDNA5 async global->LDS copy (ASYNCcnt-tracked, bypasses VGPRs) ------
// INST_OFFSET is added to BOTH the global and the LDS address (ISA 08 §4.4),
// so a linear tile copy is one address setup + N offset-immediate issues.
#if __has_builtin(__builtin_amdgcn_global_load_async_to_lds_b128)
#define ASYNC_CP16(gp, la, OFF)                                              \
  __builtin_amdgcn_global_load_async_to_lds_b128(                            \
      (gv4i_p)(uintptr_t)(gp), (lv4i_p)(uintptr_t)(la), (OFF), 0)
#else
#define ASYNC_CP16(gp, la, OFF)                                              \
  asm volatile("global_load_async_to_lds_b128 %0, %1, off offset:%2"         \
               :: "v"((unsigned)(uintptr_t)(la)),                            \
                  "v"((unsigned long long)(uintptr_t)(gp)),                  \
                  "i"(OFF) : "memory")
#endif

__device__ __forceinline__ void wait_async_le8() {
#if __has_builtin(__builtin_amdgcn_s_wait_asynccnt)
  __builtin_amdgcn_s_wait_asynccnt(8);
#else
  asm volatile("s_wait_asynccnt 0x8" ::: "memory");
#endif
}

__device__ __forceinline__ void wait_async_0() {
#if __has_builtin(__builtin_amdgcn_s_wait_asynccnt)
  __builtin_amdgcn_s_wait_asynccnt(0);
#else
  asm volatile("s_wait_asynccnt 0x0" ::: "memory");
#endif
}

// ---- native fp32 global atomic add (no-return form, no CAS loop) ----------
__device__ __forceinline__ void gatomic_fadd(float* p, float v) {
  unsigned long long a = (unsigned long long)(uintptr_t)p;
  asm volatile("global_atomic_add_f32 %0, %1, off"
               :: "v"(a), "v"(v) : "memory");
}

// ---------------------------------------------------------------------------
// Kernel: scatter-accumulate z rows into sums[label] and counts[label].
// Double-buffered 32-row tiles staged via async DMA into LDS.
// (Defined first so the disasm snippet shows the async/atomic path.)
// ---------------------------------------------------------------------------
__device__ __forceinline__ void issue_tile(const float* z, int row0, int tile,
                                           unsigned lbase, int tid) {
  // Thread owns float4 units {tid, tid+256, ...}: base addr + 8 imm offsets.
  const char* g =
      (const char*)(z + (size_t)(row0 + tile * TILE_ROWS) * EMBED_DIM) +
      (size_t)tid * 16u;
  unsigned la = lbase + (unsigned)tid * 16u;
  ASYNC_CP16(g, la, 0);
  ASYNC_CP16(g, la, 4096);
  ASYNC_CP16(g, la, 8192);
  ASYNC_CP16(g, la, 12288);
  ASYNC_CP16(g, la, 16384);
  ASYNC_CP16(g, la, 20480);
  ASYNC_CP16(g, la, 24576);
  ASYNC_CP16(g, la, 28672);
}

__global__ void __launch_bounds__(THREADS)
pmb_scatter_kernel(const float* __restrict__ z, const int* __restrict__ labels,
                   float* __restrict__ sums, float* __restrict__ counts) {
  __shared__ float buf[2][TILE_ROWS * EMBED_DIM];   // 2 * 32KB = 64KB LDS
  const int tid       = threadIdx.x;
  const int block_row = blockIdx.x * ROWS_PER_BLOCK;

  const unsigned lb0 = (unsigned)(uintptr_t)&buf[0][0];
  const unsigned lb1 = (unsigned)(uintptr_t)&buf[1][0];

  // counts: one atomic per row (ROWS_PER_BLOCK == THREADS)
  {
    int row = block_row + tid;
    int lbl = labels[row];
    gatomic_fadd(&counts[lbl], 1.0f);
  }

  // prologue: DMA tile 0 into buffer 0
  issue_tile(z, block_row, 0, lb0, tid);

  const int wave = tid >> 5;
  const int lane = tid & 31;

  for (int t = 0; t < TILES_PER_BLOCK; ++t) {
    const float* curbuf = (t & 1) ? &buf[1][0] : &buf[0][0];

    if (t + 1 < TILES_PER_BLOCK) {
      // overlap next tile's DMA with this tile's consumption
      issue_tile(z, block_row, t + 1, ((t + 1) & 1) ? lb1 : lb0, tid);
      wait_async_le8();     // oldest 8 per-wave async ops (tile t) are done
    } else {
      wait_async_0();
    }
    __syncthreads();        // all waves' portions of tile t landed in LDS

    // One wave per row. Lane owns dims {lane, lane+32, ..., lane+224}, so
    // each global_atomic_add_f32 wavefront covers a contiguous 128B segment
    // of sums[label] -> minimal L2 transactions per atomic instruction.
#pragma unroll
    for (int r = wave; r < TILE_ROWS; r += 8) {
      int row = block_row + t * TILE_ROWS + r;
      int lbl = labels[row];
      const float* srow = curbuf + r * EMBED_DIM + lane;
      float* dst = sums + (size_t)lbl * EMBED_DIM + lane;
#pragma unroll
      for (int j = 0; j < 8; ++j) {
        gatomic_fadd(dst + j * 32, srow[j * 32]);
      }
    }
    __syncthreads();        // don't let next DMA overwrite a buffer in use
  }
}

// ---------------------------------------------------------------------------
// Kernel: zero the workspace (sums[C][D] + counts[C]) every launch
// (graph replay re-runs this; d_ws is only poisoned once by the harness).
// ---------------------------------------------------------------------------
__global__ void __launch_bounds__(THREADS)
pmb_zero_kernel(float* ws, int n) {
  int i = blockIdx.x * THREADS + threadIdx.x;
  if (i < n) ws[i] = 0.0f;
}

// ---------------------------------------------------------------------------
// Kernel: finalize. One wave32 per class; lane handles 8 dims.
// mean -> l2norm -> EMA with prototypes -> l2norm -> select.
// ---------------------------------------------------------------------------
__device__ __forceinline__ float wave_reduce_sum(float v) {
#pragma unroll
  for (int off = 16; off > 0; off >>= 1)
    v += __shfl_xor(v, off, 32);
  return v;
}

__global__ void __launch_bounds__(THREADS)
pmb_finalize_kernel(const float* __restrict__ prototypes,
                    const unsigned char* __restrict__ initialized,
                    const float* __restrict__ sums,
                    const float* __restrict__ counts,
                    float* __restrict__ out) {
  const int wave = threadIdx.x >> 5;
  const int lane = threadIdx.x & 31;
  const int c = blockIdx.x * 8 + wave;
  if (c >= NUM_CLASSES) return;

  const float cnt = counts[c];
  const size_t base = (size_t)c * EMBED_DIM + lane * 8;

  const float4* sp = (const float4*)(sums + base);
  float4 s0 = sp[0], s1 = sp[1];
  const float4* pp = (const float4*)(prototypes + base);
  float4 p0 = pp[0], p1 = pp[1];

  float m[8], p[8];
  const float inv = 1.0f / fmaxf(cnt, 1.0f);
  m[0] = s0.x * inv; m[1] = s0.y * inv; m[2] = s0.z * inv; m[3] = s0.w * inv;
  m[4] = s1.x * inv; m[5] = s1.y * inv; m[6] = s1.z * inv; m[7] = s1.w * inv;
  p[0] = p0.x; p[1] = p0.y; p[2] = p0.z; p[3] = p0.w;
  p[4] = p1.x; p[5] = p1.y; p[6] = p1.z; p[7] = p1.w;

  // mean_n = l2norm(mean)
  float ss = 0.0f;
#pragma unroll
  for (int i = 0; i < 8; ++i) ss += m[i] * m[i];
  ss = wave_reduce_sum(ss);
  const float rn = 1.0f / fmaxf(sqrtf(ss), EPS_NORM);

  float mn[8], e[8];
#pragma unroll
  for (int i = 0; i < 8; ++i) mn[i] = m[i] * rn;

  // ema = l2norm(MOM*proto + (1-MOM)*mean_n)
  float ss2 = 0.0f;
#pragma unroll
  for (int i = 0; i < 8; ++i) {
    e[i] = MOMENTUM * p[i] + (1.0f - MOMENTUM) * mn[i];
    ss2 += e[i] * e[i];
  }
  ss2 = wave_reduce_sum(ss2);
  const float rn2 = 1.0f / fmaxf(sqrtf(ss2), EPS_NORM);

  const bool init    = initialized[c] != 0;
  const bool present = cnt > 0.0f;

  float o[8];
#pragma unroll
  for (int i = 0; i < 8; ++i) {
    float ema  = e[i] * rn2;
    float cand = init ? ema : mn[i];
    o[i] = present ? cand : p[i];
  }

  float4* op = (float4*)(out + base);
  op[0] = make_float4(o[0], o[1], o[2], o[3]);
  op[1] = make_float4(o[4], o[5], o[6], o[7]);
}

// ---------------------------------------------------------------------------
// Launch: inputs in setup_inputs() order: z(f32 N*D), prototypes(f32 C*D),
// initialized(bool C -> 1 byte each), labels(i32 N). Output: f32 C*D.
// ---------------------------------------------------------------------------
extern "C" void kernel_launch(void* const* d_in, const int* in_sizes, int n_in,
                              void* d_out, int out_size, void* d_ws, size_t ws_size,
                              hipStream_t stream) {
  const float*         z      = (const float*)d_in[0];
  const float*         protos = (const float*)d_in[1];
  const unsigned char* init   = (const unsigned char*)d_in[2];
  const int*           labels = (const int*)d_in[3];
  float*               out    = (float*)d_out;

  float* sums   = (float*)d_ws;                     // [C][D]
  float* counts = sums + NUM_CLASSES * EMBED_DIM;   // [C]

  const int N = in_sizes[3];                        // 131072

  const int nzero = NUM_CLASSES * EMBED_DIM + NUM_CLASSES;
  pmb_zero_kernel<<<(nzero + THREADS - 1) / THREADS, THREADS, 0, stream>>>(sums, nzero);

  const int nblocks = N / ROWS_PER_BLOCK;           // 512
  pmb_scatter_kernel<<<nblocks, THREADS, 0, stream>>>(z, labels, sums, counts);

  pmb_finalize_kernel<<<(NUM_CLASSES + 7) / 8, THREADS, 0, stream>>>(
      protos, init, sums, counts, out);
}